// PairwiseModel_76416058131343
// MI455X (gfx1250) — compile-verified
//
#include <hip/hip_runtime.h>
#include <hip/hip_bf16.h>

typedef __attribute__((ext_vector_type(2))) float v2f;
typedef __attribute__((ext_vector_type(8))) float v8f;
typedef __attribute__((ext_vector_type(4))) unsigned int u32x4;
typedef __attribute__((ext_vector_type(4))) int i32x4;
typedef __attribute__((ext_vector_type(8))) int i32x8;

#define NB 8192
#define DD 128
#define NTILES 512            // NB / 16
#define BLK 64                // pairwise block edge (rows and cols)
#define NBLK (NB / BLK)       // 128

// ---------------------------------------------------------------------------
// TDM: issue a tensor_load_to_lds of a (rows x 128) f32 tile of h into LDS.
// D# layout per CDNA5 ISA §8.3/8.4:
//   group0: count=1 | lds_addr | global_addr[56:0] | type=2
//   group1: data_size=4B, tensor_dim0=128, tensor_dim1=8192,
//           tile_dim0=128, tile_dim1=rows, dim0_stride=128
// Wave-level op (EXEC ignored); tracked by TENSORcnt.
// ---------------------------------------------------------------------------
__device__ __forceinline__ void tdm_load_tile(const float* gptr,
                                              unsigned lds_off, int rows) {
#if __has_builtin(__builtin_amdgcn_tensor_load_to_lds)
  const unsigned long long ga = (unsigned long long)(uintptr_t)gptr;
  u32x4 g0;
  g0[0] = 1u;                                          // count=1, user D#
  g0[1] = lds_off;                                     // lds_addr (bytes)
  g0[2] = (unsigned)(ga & 0xffffffffu);                // global_addr[31:0]
  g0[3] = (unsigned)((ga >> 32) & 0x01ffffffu) | (2u << 30); // [56:32]|type=2
  i32x8 g1;
  g1[0] = (int)(2u << 16);                   // wg_mask=0, data_size=2 (4B)
  g1[1] = (int)(128u << 16);                 // tensor_dim0 = 128
  g1[2] = (int)((8192u & 0xffffu) << 16);    // tensor_dim1.lo = 8192
  g1[3] = (int)(128u << 16);                 // tensor_dim1.hi=0 | tile_dim0=128
  g1[4] = rows;                              // tile_dim1 = rows, tile_dim2=0
  g1[5] = 128;                               // tensor_dim0_stride = 128
  g1[6] = 0;
  g1[7] = 0;
  i32x4 z4 = {0, 0, 0, 0};                   // groups 2/3 unused (2D tensor)
#if defined(__clang_major__) && (__clang_major__ >= 23)
  i32x8 z8 = {0, 0, 0, 0, 0, 0, 0, 0};
  __builtin_amdgcn_tensor_load_to_lds(g0, g1, z4, z4, z8, 0);
#else
  __builtin_amdgcn_tensor_load_to_lds(g0, g1, z4, z4, 0);
#endif
#endif
}

// ---------------------------------------------------------------------------
// Kernel 1: fused 3-layer MLP.  One wave per 16-row tile of X; activations
// ping-pong in LDS, weights through WGP$/L2.  32 x V_WMMA_F32_16X16X4_F32
// per 16x16 output tile.
// ---------------------------------------------------------------------------
__global__ __launch_bounds__(32) void mlp_kernel(
    const float* __restrict__ X,
    const float* __restrict__ W1, const float* __restrict__ b1,
    const float* __restrict__ W2, const float* __restrict__ b2,
    const float* __restrict__ W3, const float* __restrict__ b3,
    float* __restrict__ h, float* __restrict__ sq) {
  __shared__ float buf0[16 * DD];
  __shared__ float buf1[16 * DD];

  const int lane = threadIdx.x;          // wave32
  const int row0 = blockIdx.x * 16;
  const int m  = lane & 15;              // row (A) / col (B,D) index
  const int kk = lane >> 4;              // K-half select

  {
    const float4* src = (const float4*)(X + (size_t)row0 * DD);
    float4* dst = (float4*)buf0;
    for (int t = lane; t < 512; t += 32) dst[t] = src[t];
  }
  __syncthreads();

  const float* Ws[3]   = {W1, W2, W3};
  const float* bias[3] = {b1, b2, b3};
  float* in  = buf0;
  float* out = buf1;

  for (int layer = 0; layer < 3; ++layer) {
    const float* W = Ws[layer];
    const float* bv = bias[layer];
    const bool relu = (layer < 2);
    for (int nt = 0; nt < 8; ++nt) {
      const int n0 = nt * 16;
      const float bb = bv[n0 + m];
      v8f c = {bb, bb, bb, bb, bb, bb, bb, bb};
      #pragma unroll 8
      for (int k = 0; k < DD; k += 4) {
        const float* ap = in + m * DD + k + 2 * kk;
        v2f a; a.x = ap[0]; a.y = ap[1];
        const float* bp = W + (size_t)(k + 2 * kk) * DD + n0 + m;
        v2f bf; bf.x = bp[0]; bf.y = bp[DD];
        c = __builtin_amdgcn_wmma_f32_16x16x4_f32(
            false, a, false, bf, (short)0, c, false, false);
      }
      #pragma unroll
      for (int r = 0; r < 8; ++r) {
        float v = c[r];
        if (relu) v = fmaxf(v, 0.0f);
        out[(r + 8 * kk) * DD + n0 + m] = v;
      }
    }
    __syncthreads();
    float* t = in; in = out; out = t;
  }

  {
    float4* dst = (float4*)(h + (size_t)row0 * DD);
    const float4* src = (const float4*)in;
    for (int t = lane; t < 512; t += 32) dst[t] = src[t];
  }
  float s = 0.0f;
  const float* rowp = in + m * DD + kk * 64;
  #pragma unroll 8
  for (int c2 = 0; c2 < 64; ++c2) { const float v = rowp[c2]; s += v * v; }
  s += __shfl_xor(s, 16, 32);
  if (lane < 16) sq[row0 + lane] = s;
}

// ---------------------------------------------------------------------------
// Kernel 2: pairwise loss with TDM-staged LDS tiles.
// Block (256 thr = 8 waves) owns a 64x64 output block; block-level triangular
// symmetry (bj>bi weighted x2).  Wave 0 issues two tensor_load_to_lds
// (64x128 f32 tiles of h), s_wait_tensorcnt, barrier; each wave then computes
// 2 of the 16 (16x16) tiles: 64 WMMA from ds_load fragments + fused exp/log.
// ---------------------------------------------------------------------------
__global__ __launch_bounds__(256) void pair_kernel(
    const float* __restrict__ h, const float* __restrict__ sq,
    const int* __restrict__ y, float* __restrict__ acc) {
  __shared__ float Ah[BLK * DD];     // 32KB
  __shared__ float Bh[BLK * DD];     // 32KB

  const int bi = blockIdx.x / NBLK;
  const int bj = blockIdx.x % NBLK;
  if (bj < bi) return;                       // block-uniform: EXEC stays full
  const float weight = (bj > bi) ? 2.0f : 1.0f;
  const int i0 = bi * BLK, j0 = bj * BLK;

  const int lane = threadIdx.x & 31;
  const int wv   = threadIdx.x >> 5;

#if __has_builtin(__builtin_amdgcn_tensor_load_to_lds)
  if (wv == 0) {                             // one wave drives the TDM
    tdm_load_tile(h + (size_t)i0 * DD, (unsigned)(uintptr_t)&Ah[0], BLK);
    tdm_load_tile(h + (size_t)j0 * DD, (unsigned)(uintptr_t)&Bh[0], BLK);
    __builtin_amdgcn_s_wait_tensorcnt(0);
  }
#else
  for (int t = threadIdx.x; t < BLK * DD / 4; t += 256) {
    ((float4*)Ah)[t] = ((const float4*)(h + (size_t)i0 * DD))[t];
    ((float4*)Bh)[t] = ((const float4*)(h + (size_t)j0 * DD))[t];
  }
#endif
  __syncthreads();                           // publish LDS to all 8 waves

  const int m  = lane & 15;
  const int kk = lane >> 4;
  float contrib = 0.0f;

  #pragma unroll
  for (int sub = 0; sub < 2; ++sub) {        // 16 tiles / 8 waves
    const int t  = wv * 2 + sub;
    const int tr = t >> 2, tc = t & 3;       // 4x4 tile grid in the block
    const float* Ap = Ah + (tr * 16 + m) * DD + 2 * kk;
    const float* Bp = Bh + (tc * 16 + m) * DD + 2 * kk;

    v8f c = {0, 0, 0, 0, 0, 0, 0, 0};
    #pragma unroll 8
    for (int k = 0; k < DD; k += 4) {
      v2f a = *(const v2f*)(Ap + k);         // ds_load_b64
      v2f b = *(const v2f*)(Bp + k);
      c = __builtin_amdgcn_wmma_f32_16x16x4_f32(
          false, a, false, b, (short)0, c, false, false);
    }

    const int jbase = j0 + tc * 16 + m;      // N = m for all held elements
    const float sqj = sq[jbase];
    const int   yj  = y[jbase];
    #pragma unroll
    for (int r = 0; r < 8; ++r) {
      const int ii = i0 + tr * 16 + r + 8 * kk;   // M = r + 8*kk
      float d = sq[ii] + sqj - 2.0f * c[r];
      d = fmaxf(d, 0.0f);
      contrib += (y[ii] != yj) ? -__logf(1.01f - __expf(-d)) : d;
    }
  }

  #pragma unroll
  for (int off = 16; off > 0; off >>= 1) contrib += __shfl_xor(contrib, off, 32);
  if (lane == 0) atomicAdd(acc, contrib * weight);
}

// ---------------------------------------------------------------------------
// Kernel 3: normalize + regularizer.
// ---------------------------------------------------------------------------
__global__ __launch_bounds__(256) void finalize_kernel(
    const float* __restrict__ acc, const float* __restrict__ sq,
    float* __restrict__ out) {
  __shared__ float red[256];
  float s = 0.0f;
  for (int i = threadIdx.x; i < NB; i += 256) s += sq[i];
  red[threadIdx.x] = s;
  __syncthreads();
  for (int off = 128; off > 0; off >>= 1) {
    if (threadIdx.x < off) red[threadIdx.x] += red[threadIdx.x + off];
    __syncthreads();
  }
  if (threadIdx.x == 0) {
    const float inv_bb = 1.0f / ((float)NB * (float)NB);
    const float inv_bd = 1.0f / ((float)NB * (float)DD);
    out[0] = acc[0] * inv_bb + 0.01f * red[0] * inv_bd;
  }
}

extern "C" void kernel_launch(void* const* d_in, const int* in_sizes, int n_in,
                              void* d_out, int out_size, void* d_ws, size_t ws_size,
                              hipStream_t stream) {
  const float* X  = (const float*)d_in[0];
  const float* W1 = (const float*)d_in[1];
  const float* b1 = (const float*)d_in[2];
  const float* W2 = (const float*)d_in[3];
  const float* b2 = (const float*)d_in[4];
  const float* W3 = (const float*)d_in[5];
  const float* b3 = (const float*)d_in[6];
  const int*   y  = (const int*)d_in[7];
  float* out = (float*)d_out;

  float* acc = (float*)d_ws;
  float* h   = (float*)((char*)d_ws + 256);
  float* sq  = h + (size_t)NB * DD;

  hipMemsetAsync(d_ws, 0, 256, stream);
  mlp_kernel<<<NTILES, 32, 0, stream>>>(X, W1, b1, W2, b2, W3, b3, h, sq);
  pair_kernel<<<NBLK * NBLK, 256, 0, stream>>>(h, sq, y, acc);
  finalize_kernel<<<1, 256, 0, stream>>>(acc, sq, out);
}